// RNNWithFixedWeights_8589935288
// MI455X (gfx1250) — compile-verified
//
#include <hip/hip_runtime.h>
#include <hip/hip_bf16.h>

// ---------------------------------------------------------------------------
// Packed-sequence RNN (H=3) scan for MI455X / gfx1250.
//
// 1 lane = 1 sequence, 1 wave32 = 1 workgroup = 32 consecutive sequences.
// Load addresses are independent of the recurrence, so each wave runs a
// depth-16 pipeline of CDNA5 async global->LDS copies: ONE b128 async per
// step moves the wave's 384B slice (lanes 0..23 x 16B, tail lanes clamp onto
// the last window).  Wave-uniform bookkeeping is scalarized via
// readfirstlane; s_wait_asynccnt retires only the oldest slot.
// ---------------------------------------------------------------------------

#define PIPE_DEPTH 16  // steps in flight

typedef int v4i __attribute__((vector_size(16)));
typedef __attribute__((address_space(1))) int gint_t;   // global int
typedef __attribute__((address_space(3))) int lint_t;   // LDS int
typedef __attribute__((address_space(1))) v4i gv4i_t;   // global int4
typedef __attribute__((address_space(3))) v4i lv4i_t;   // LDS int4

__device__ __forceinline__ void async_b32(const void* g, void* l) {
#if __has_builtin(__builtin_amdgcn_global_load_async_to_lds_b32)
  __builtin_amdgcn_global_load_async_to_lds_b32((gint_t*)g, (lint_t*)l, 0, 0);
#else
  *(volatile int*)l = *(const int*)g;  // sync fallback
#endif
}

#if __has_builtin(__builtin_amdgcn_global_load_async_to_lds_b128)
#define ASYNCS_PER_STEP 1
__device__ __forceinline__ void async_step16(const void* g, void* l) {
  __builtin_amdgcn_global_load_async_to_lds_b128((gv4i_t*)g, (lv4i_t*)l, 0, 0);
}
#else
#define ASYNCS_PER_STEP 4
__device__ __forceinline__ void async_step16(const void* g, void* l) {
#pragma unroll
  for (int q = 0; q < 4; ++q)
    async_b32((const char*)g + 4 * q, (char*)l + 4 * q);
}
#endif

__device__ __forceinline__ void wait_async0() {
#if __has_builtin(__builtin_amdgcn_s_wait_asynccnt)
  __builtin_amdgcn_s_wait_asynccnt(0);
#else
  asm volatile("s_wait_asynccnt 0" ::: "memory");
#endif
}

__device__ __forceinline__ void wait_async_pipe() {
#if __has_builtin(__builtin_amdgcn_s_wait_asynccnt)
  __builtin_amdgcn_s_wait_asynccnt(ASYNCS_PER_STEP * (PIPE_DEPTH - 1));
#else
  asm volatile("s_wait_asynccnt 45" ::: "memory");
#endif
}

__device__ __forceinline__ void wait_ds0() {
#if __has_builtin(__builtin_amdgcn_s_wait_dscnt)
  __builtin_amdgcn_s_wait_dscnt(0);
#else
  asm volatile("s_wait_dscnt 0" ::: "memory");
#endif
}

// inv[unsort[j]] = j  (final-state scatter permutation)
__global__ void build_inverse_kernel(const int* __restrict__ unsort,
                                     int* __restrict__ inv, int B) {
  int j = blockIdx.x * blockDim.x + threadIdx.x;
  if (j < B) inv[unsort[j]] = j;
}

__global__ __launch_bounds__(32, 1)
void rnn_packed_scan_kernel(const float* __restrict__ data,
                            const int* __restrict__ batch_sizes,
                            const float* __restrict__ w1p,
                            const float* __restrict__ w2p,
                            const int* __restrict__ inv,
                            float* __restrict__ out,
                            int T, int B, int totBytes) {
  extern __shared__ int smem[];
  const int Talign = (T + 3) & ~3;       // keep xbuf 16B-aligned
  int* bs_lds = smem;                    // T ints (batch_sizes table)
  int* xbuf   = smem + Talign;           // PIPE_DEPTH slots * 96 dwords (384B)
  float* xf   = (float*)xbuf;

  const int lane  = threadIdx.x;         // 0..31
  const int wbase = blockIdx.x * 32;     // first sequence of this wave
  const int b     = wbase + lane;        // this lane's sequence

  // ---- one-time async staging of batch_sizes into LDS ----
  for (int i = lane; i < T; i += 32)
    async_b32(batch_sizes + i, bs_lds + i);
  wait_async0();

  // ---- per-lane sequence length: first t with bs[t] <= b (bs non-increasing)
  int lo = 0, hi = T;
  while (lo < hi) {
    int m = (lo + hi) >> 1;
    if (bs_lds[m] > b) lo = m + 1; else hi = m;
  }
  const int mylen = lo;
  const int wlen  = __builtin_amdgcn_readfirstlane(mylen);  // lane 0 = longest

  const float w1 = w1p[0];
  const float w2 = w2p[0];

  // per-lane constant: byte window inside the 384B step slice
  const int sl   = (lane < 24) ? lane * 16 : 368;  // lanes 24..31 dup the tail
  const int gmax = totBytes - 16;                  // clamp for end-of-array

  // ---- async pipeline state (wave-uniform -> SALU via readfirstlane) ----
  int pref_rowB   = wbase * 12;  // byte addr of slice start at step pref_t
  int pref_t      = 0;
  int frozen_rowB = pref_rowB;   // reused for dummy issues past wlen (keeps
                                 // ASYNCcnt arithmetic exact, EXEC never 0)

  auto issue_next = [&]() {
    int rowB;
    if (pref_t < wlen) {
      int bsv = __builtin_amdgcn_readfirstlane(bs_lds[pref_t]);
      rowB        = pref_rowB;
      frozen_rowB = rowB;
      pref_rowB  += bsv * 12;
    } else {
      rowB = frozen_rowB;        // dummy: same issue count, data unused
    }
    int gbyte = rowB + sl;
    gbyte     = gbyte < gmax ? gbyte : gmax;   // stay inside data[]
    const char* gp = (const char*)data + gbyte;
    char*       lp = (char*)xbuf + ((pref_t & (PIPE_DEPTH - 1)) * 384 + sl);
    async_step16(gp, lp);
    ++pref_t;
  };

  // ---- preload PIPE_DEPTH steps ----
#pragma unroll
  for (int p = 0; p < PIPE_DEPTH; ++p) issue_next();

  // ---- serial scan: retire one slot / issue one slot per step ----
  float h0 = 0.f, h1 = 0.f, h2 = 0.f;
  for (int t = 0; t < wlen; ++t) {
    wait_async_pipe();                    // oldest slot landed in LDS
    const int sb = (t & (PIPE_DEPTH - 1)) * 96 + lane * 3;
    float x0 = xf[sb + 0];
    float x1 = xf[sb + 1];
    float x2 = xf[sb + 2];
    if (t < mylen) {
      float n0 = fmaxf(0.f, __builtin_fmaf(w1, h0, x0));
      float n1 = fmaxf(0.f, __builtin_fmaf(w2, h0, (x1 + h1) - h2));
      float n2 = fmaxf(0.f, x2 + h0);
      h0 = n0; h1 = n1; h2 = n2;
    }
    wait_ds0();                           // slot reads done before overwrite
    issue_next();                         // refill with step t+PIPE_DEPTH
  }
  // trailing dummy asyncs are flushed by S_ENDPGM's implicit wait-idle

  if (b < B) {
    int j = inv[b];
    out[3 * j + 0] = h0;
    out[3 * j + 1] = h1;
    out[3 * j + 2] = h2;
  }
}

extern "C" void kernel_launch(void* const* d_in, const int* in_sizes, int n_in,
                              void* d_out, int out_size, void* d_ws, size_t ws_size,
                              hipStream_t stream) {
  const float* data        = (const float*)d_in[0];
  const int*   batch_sizes = (const int*)d_in[1];
  const int*   unsort      = (const int*)d_in[2];
  const float* w1          = (const float*)d_in[3];
  const float* w2          = (const float*)d_in[4];
  const int T        = in_sizes[1];        // timesteps (2048)
  const int B        = in_sizes[2];        // sequences (8192)
  const int totBytes = in_sizes[0] * 4;    // bytes of packed data

  float* out = (float*)d_out;
  int*   inv = (int*)d_ws;                 // B ints of scratch

  build_inverse_kernel<<<(B + 255) / 256, 256, 0, stream>>>(unsort, inv, B);

  const int    nblocks = (B + 31) / 32;    // one wave32 per block
  const size_t shmem   = (size_t)((T + 3) & ~3) * sizeof(int)
                       + (size_t)PIPE_DEPTH * 384;
  rnn_packed_scan_kernel<<<nblocks, 32, shmem, stream>>>(
      data, batch_sizes, w1, w2, inv, out, T, B, totBytes);
}